// TowerGNN_38594576122130
// MI455X (gfx1250) — compile-verified
//
#include <hip/hip_runtime.h>
#include <hip/hip_bf16.h>

// TowerGNN on MI455X (gfx1250, wave32).
// GEMMs via V_WMMA_F32_16X16X4_F32 (fp32-exact); edge aggregation via
// global_atomic_add_f32 with float4 gathers (L2-resident feature matrices).

#define EPS 1e-5f

typedef __attribute__((ext_vector_type(2))) float v2f;
typedef __attribute__((ext_vector_type(8))) float v8f;

// ---------------------------------------------------------------- degrees ---
__global__ void deg_init(float* __restrict__ deg, int n) {
    int i = blockIdx.x * blockDim.x + threadIdx.x;
    if (i < n) deg[i] = 1.0f;                 // self-loop contributes 1
}

__global__ void deg_edges(const int* __restrict__ dst, float* __restrict__ deg, int e) {
    int i = blockIdx.x * blockDim.x + threadIdx.x;
    if (i < e) atomicAdd(&deg[dst[i]], 1.0f);
}

__global__ void deg_to_dinv(float* __restrict__ deg, int n) {
    int i = blockIdx.x * blockDim.x + threadIdx.x;
    if (i < n) {
        float d = deg[i];
        deg[i] = (d > 0.0f) ? __frsqrt_rn(d) : 0.0f;
    }
}

// ------------------------------------------------------------- WMMA GEMM ----
// H[n,64] = X[n,FIN] @ W[FIN,64].  One wave = one 16x16 tile of H.
// Block = 128 threads = 4 waves covering the 4 column tiles of one row slab.
// n must be a multiple of 16 (holds: 100000, 256) so EXEC is all ones.
template <int FIN>
__global__ __launch_bounds__(128) void gemm_wmma(const float* __restrict__ X,
                                                 const float* __restrict__ W,
                                                 float* __restrict__ H, int n) {
    const int lane = threadIdx.x & 31;
    const int wave = threadIdx.x >> 5;          // column tile 0..3
    const int row0 = blockIdx.x * 16;
    const int n0   = wave * 16;
    const int m    = lane & 15;                 // A row within tile
    const int nl   = lane & 15;                 // B/C column within tile
    const int kb   = (lane >> 4) * 2;           // K sub-offset per half-wave

    const float* xr = X + (size_t)(row0 + m) * FIN;

    v8f acc = {};
#pragma unroll
    for (int k0 = 0; k0 < FIN; k0 += 4) {
        v2f a, b;
        a.x = xr[k0 + kb];
        a.y = xr[k0 + kb + 1];
        b.x = W[(size_t)(k0 + kb) * 64 + n0 + nl];
        b.y = W[(size_t)(k0 + kb + 1) * 64 + n0 + nl];
        acc = __builtin_amdgcn_wmma_f32_16x16x4_f32(
            /*neg_a=*/false, a, /*neg_b=*/false, b,
            /*c_mod=*/(short)0, acc, /*reuse_a=*/false, /*reuse_b=*/false);
    }

    const int rbase = row0 + 8 * (lane >> 4);
#pragma unroll
    for (int v = 0; v < 8; ++v)
        H[(size_t)(rbase + v) * 64 + n0 + nl] = acc[v];
}

// ------------------------------------------------------------ aggregation ---
// ACC[i,:] = dinv[i]^2 * H[i,:]   (self-loop term; one thread per 4 features)
__global__ void agg_init(const float* __restrict__ H, const float* __restrict__ dinv,
                         float* __restrict__ acc, int n) {
    int t = blockIdx.x * blockDim.x + threadIdx.x;
    if (t >= n * 16) return;
    int i = t >> 4;
    int c = (t & 15) * 4;
    float w = dinv[i];
    w *= w;
    float4 h = *(const float4*)(H + (size_t)i * 64 + c);
    float4 o = make_float4(w * h.x, w * h.y, w * h.z, w * h.w);
    *(float4*)(acc + (size_t)i * 64 + c) = o;
}

// ACC[dst,:] += dinv[src]*dinv[dst] * H[src,:]   (16 threads per edge)
__global__ void edge_scatter(const float* __restrict__ H, const int* __restrict__ src,
                             const int* __restrict__ dst, const float* __restrict__ dinv,
                             float* __restrict__ acc, int e) {
    long long t = (long long)blockIdx.x * blockDim.x + threadIdx.x;
    long long ei = t >> 4;
    if (ei >= e) return;
    int c = (int)(t & 15) * 4;
    int s = src[ei], d = dst[ei];
    float w = dinv[s] * dinv[d];
    float4 h = *(const float4*)(H + (size_t)s * 64 + c);
    float* p = acc + (size_t)d * 64 + c;
    atomicAdd(p + 0, w * h.x);
    atomicAdd(p + 1, w * h.y);
    atomicAdd(p + 2, w * h.z);
    atomicAdd(p + 3, w * h.w);
}

// x = relu(((acc + b) - rm) * rsqrt(rv+eps) * g + be), in place
__global__ void bn_relu(float* __restrict__ acc, const float* __restrict__ b,
                        const float* __restrict__ g, const float* __restrict__ be,
                        const float* __restrict__ rm, const float* __restrict__ rv,
                        int n) {
    int t = blockIdx.x * blockDim.x + threadIdx.x;
    if (t >= n * 64) return;
    int f = t & 63;
    float v = (acc[t] + b[f] - rm[f]) * __frsqrt_rn(rv[f] + EPS) * g[f] + be[f];
    acc[t] = fmaxf(v, 0.0f);
}

// ---------------------------------------------------------------- pooling ---
__global__ void pool_init(float* __restrict__ sums, float* __restrict__ cnts, int gn) {
    int t = blockIdx.x * blockDim.x + threadIdx.x;
    if (t < gn * 64) sums[t] = 0.0f;
    if (t < gn) cnts[t] = 0.0f;
}

__global__ void pool_scatter(const float* __restrict__ h, const int* __restrict__ batch,
                             float* __restrict__ sums, float* __restrict__ cnts, int n) {
    int t = blockIdx.x * blockDim.x + threadIdx.x;
    if (t >= n * 16) return;
    int i = t >> 4;
    int c = (t & 15) * 4;
    int g = batch[i];
    float4 v = *(const float4*)(h + (size_t)i * 64 + c);
    float* p = sums + (size_t)g * 64 + c;
    atomicAdd(p + 0, v.x);
    atomicAdd(p + 1, v.y);
    atomicAdd(p + 2, v.z);
    atomicAdd(p + 3, v.w);
    if (c == 0) atomicAdd(&cnts[g], 1.0f);
}

__global__ void pool_finalize(float* __restrict__ sums, const float* __restrict__ cnts, int gn) {
    int t = blockIdx.x * blockDim.x + threadIdx.x;
    if (t >= gn * 64) return;
    sums[t] /= fmaxf(cnts[t >> 6], 1.0f);
}

// ------------------------------------------------------------------- head ---
__global__ void bias_relu(float* __restrict__ h, const float* __restrict__ b, int gn) {
    int t = blockIdx.x * blockDim.x + threadIdx.x;
    if (t >= gn * 64) return;
    h[t] = fmaxf(h[t] + b[t & 63], 0.0f);
}

__global__ void head2(const float* __restrict__ hid, const float* __restrict__ W,
                      const float* __restrict__ b, float* __restrict__ out, int gn) {
    int t = blockIdx.x * blockDim.x + threadIdx.x;
    if (t >= gn * 2) return;
    int g = t >> 1, c = t & 1;
    const float* hr = hid + (size_t)g * 64;
    float s = 0.0f;
#pragma unroll
    for (int j = 0; j < 64; ++j) s += hr[j] * W[j * 2 + c];
    out[t] = s + b[c];
}

// ----------------------------------------------------------------- driver ---
extern "C" void kernel_launch(void* const* d_in, const int* in_sizes, int n_in,
                              void* d_out, int out_size, void* d_ws, size_t ws_size,
                              hipStream_t stream) {
    const float* x     = (const float*)d_in[0];
    const int*   ei    = (const int*)d_in[1];
    const int*   batch = (const int*)d_in[2];
    const float* W1 = (const float*)d_in[3],  *b1 = (const float*)d_in[4];
    const float* g1 = (const float*)d_in[5],  *be1 = (const float*)d_in[6];
    const float* rm1 = (const float*)d_in[7], *rv1 = (const float*)d_in[8];
    const float* W2 = (const float*)d_in[9],  *b2 = (const float*)d_in[10];
    const float* g2 = (const float*)d_in[11], *be2 = (const float*)d_in[12];
    const float* rm2 = (const float*)d_in[13], *rv2 = (const float*)d_in[14];
    const float* W3 = (const float*)d_in[15], *b3 = (const float*)d_in[16];
    const float* g3 = (const float*)d_in[17], *be3 = (const float*)d_in[18];
    const float* rm3 = (const float*)d_in[19], *rv3 = (const float*)d_in[20];
    const float* Wr1 = (const float*)d_in[21], *br1 = (const float*)d_in[22];
    const float* Wr2 = (const float*)d_in[23], *br2 = (const float*)d_in[24];

    const int N  = in_sizes[0] / 8;     // 100000 (multiple of 16)
    const int E  = in_sizes[1] / 2;     // 3200000
    const int Gn = 256;

    const int* src = ei;
    const int* dst = ei + E;

    // workspace carve-up (floats): dinv | H | ACC | sums | cnts | hid
    float* ws   = (float*)d_ws;
    float* dinv = ws;
    float* H    = dinv + N;
    float* ACC  = H + (size_t)N * 64;
    float* sums = ACC + (size_t)N * 64;
    float* cnts = sums + (size_t)Gn * 64;
    float* hid  = cnts + Gn;

    const int TB = 256;
    auto blocks = [](long long work, int tb) { return (int)((work + tb - 1) / tb); };

    // symmetric normalization: dinv = deg^-1/2 (self-loops folded in)
    deg_init<<<blocks(N, TB), TB, 0, stream>>>(dinv, N);
    deg_edges<<<blocks(E, TB), TB, 0, stream>>>(dst, dinv, E);
    deg_to_dinv<<<blocks(N, TB), TB, 0, stream>>>(dinv, N);

    // ---- layer 1 (8 -> 64) ----
    gemm_wmma<8><<<N / 16, 128, 0, stream>>>(x, W1, H, N);
    agg_init<<<blocks((long long)N * 16, TB), TB, 0, stream>>>(H, dinv, ACC, N);
    edge_scatter<<<blocks((long long)E * 16, TB), TB, 0, stream>>>(H, src, dst, dinv, ACC, E);
    bn_relu<<<blocks((long long)N * 64, TB), TB, 0, stream>>>(ACC, b1, g1, be1, rm1, rv1, N);

    // ---- layer 2 (64 -> 64) ----
    gemm_wmma<64><<<N / 16, 128, 0, stream>>>(ACC, W2, H, N);
    agg_init<<<blocks((long long)N * 16, TB), TB, 0, stream>>>(H, dinv, ACC, N);
    edge_scatter<<<blocks((long long)E * 16, TB), TB, 0, stream>>>(H, src, dst, dinv, ACC, E);
    bn_relu<<<blocks((long long)N * 64, TB), TB, 0, stream>>>(ACC, b2, g2, be2, rm2, rv2, N);

    // ---- layer 3 (64 -> 64) ----
    gemm_wmma<64><<<N / 16, 128, 0, stream>>>(ACC, W3, H, N);
    agg_init<<<blocks((long long)N * 16, TB), TB, 0, stream>>>(H, dinv, ACC, N);
    edge_scatter<<<blocks((long long)E * 16, TB), TB, 0, stream>>>(H, src, dst, dinv, ACC, E);
    bn_relu<<<blocks((long long)N * 64, TB), TB, 0, stream>>>(ACC, b3, g3, be3, rm3, rv3, N);

    // ---- global mean pool ----
    pool_init<<<blocks((long long)Gn * 64, TB), TB, 0, stream>>>(sums, cnts, Gn);
    pool_scatter<<<blocks((long long)N * 16, TB), TB, 0, stream>>>(ACC, batch, sums, cnts, N);
    pool_finalize<<<blocks((long long)Gn * 64, TB), TB, 0, stream>>>(sums, cnts, Gn);

    // ---- MLP head: relu(pooled @ Wr1 + br1) @ Wr2 + br2 ----
    gemm_wmma<64><<<Gn / 16, 128, 0, stream>>>(sums, Wr1, hid, Gn);
    bias_relu<<<blocks((long long)Gn * 64, TB), TB, 0, stream>>>(hid, br1, Gn);
    head2<<<blocks((long long)Gn * 2, TB), TB, 0, stream>>>(hid, Wr2, br2, (float*)d_out, Gn);
}